// DistGCN_90357521973889
// MI455X (gfx1250) — compile-verified
//
#include <hip/hip_runtime.h>
#include <hip/hip_bf16.h>
#include <cstdint>
#include <cstddef>

// GCN 2-layer forward for MI455X (gfx1250, wave32).
//   h0 = x@W0+b0 ; a0 = Ahat@h0 ; h1 = relu(a0)@W1+b1 ; a1 = Ahat@h1 ; out = log_softmax(a1)
// GEMMs use v_wmma_f32_16x16x4_f32 (fp32 path; GEMM flops are negligible vs the
// L2-resident edge aggregation, so no precision downcast is needed).

typedef float v2f __attribute__((ext_vector_type(2)));
typedef float v8f __attribute__((ext_vector_type(8)));

#define FDIM 128

// ---------------- degree / dinv ----------------
__global__ void k_init_deg(float* __restrict__ deg, int N) {
    int i = blockIdx.x * blockDim.x + threadIdx.x;
    if (i < N) deg[i] = 1.0f;               // self loop
}

__global__ void k_count_deg(const int* __restrict__ col, float* __restrict__ deg, int E) {
    int e = blockIdx.x * blockDim.x + threadIdx.x;
    if (e < E) atomicAdd(&deg[col[e]], 1.0f);
}

__global__ void k_finalize_dinv(float* __restrict__ deg, int N) {
    int i = blockIdx.x * blockDim.x + threadIdx.x;
    if (i < N) deg[i] = rsqrtf(deg[i]);
}

// ---------------- WMMA GEMM: out[N,128] = (relu?)X[N,128] @ W[128,128] + b ----------------
// 128 threads = 4 waves; wave w computes rows [blk*64 + w*16, +16) x all 128 cols.
// A layout (32-bit 16x4): lanes 0-15 hold M=lane, K={k,k+1}; lanes 16-31 hold K={k+2,k+3}.
// B layout (4x16, mirrored): lane = 16*(kk>=2) + n, vgpr = kk&1.
// C/D layout: VGPR v, lane L -> M = v + 8*(L>=16), N = L%16.
template<bool RELU_IN>
__global__ __launch_bounds__(128) void k_gemm_bias(const float* __restrict__ X,
                                                   const float* __restrict__ W,
                                                   const float* __restrict__ bias,
                                                   float* __restrict__ out, int N) {
    __shared__ float Wsh[FDIM * FDIM];      // 64 KB (row-major W)

    int tid = threadIdx.x;
    // cooperative, coalesced W -> LDS (float4)
    const float4* Wv = (const float4*)W;
    float4*       Sv = (float4*)Wsh;
#pragma unroll
    for (int i = 0; i < (FDIM * FDIM / 4) / 128; ++i)
        Sv[tid + i * 128] = Wv[tid + i * 128];
    __syncthreads();

    const int lane  = tid & 31;
    const int wave  = tid >> 5;
    const int m16   = lane & 15;
    const int khalf = lane >> 4;            // 0 or 1
    const int rbase = blockIdx.x * 64 + wave * 16;
    const int rowA  = rbase + m16;
    const bool rvalid = rowA < N;
    const float* xrow = X + (size_t)rowA * FDIM;

    // register-resident A fragments for the whole K=128 (32 k-steps of 4)
    v2f a[32];
#pragma unroll
    for (int ks = 0; ks < 32; ++ks) {
        float ax = 0.0f, ay = 0.0f;
        if (rvalid) {
            int kk = ks * 4 + 2 * khalf;
            ax = xrow[kk];
            ay = xrow[kk + 1];
            if (RELU_IN) { ax = fmaxf(ax, 0.0f); ay = fmaxf(ay, 0.0f); }
        }
        a[ks][0] = ax; a[ks][1] = ay;
    }

#pragma unroll
    for (int nt = 0; nt < 8; ++nt) {
        const int n = nt * 16 + m16;
        v8f c = {0.0f, 0.0f, 0.0f, 0.0f, 0.0f, 0.0f, 0.0f, 0.0f};
#pragma unroll
        for (int ks = 0; ks < 32; ++ks) {
            const int kk = ks * 4 + 2 * khalf;
            v2f b;
            b[0] = Wsh[kk * FDIM + n];
            b[1] = Wsh[(kk + 1) * FDIM + n];
            // D = A(16x4) * B(4x16) + C  -> v_wmma_f32_16x16x4_f32
            c = __builtin_amdgcn_wmma_f32_16x16x4_f32(
                    false, a[ks], false, b, (short)0, c, false, false);
        }
        const float bv = bias[n];
#pragma unroll
        for (int v = 0; v < 8; ++v) {
            const int r = rbase + v + 8 * khalf;
            if (r < N) out[(size_t)r * FDIM + n] = c[v] + bv;
        }
    }
}

// ---------------- aggregation ----------------
// agg[i,:] = dinv[i]^2 * h[i,:]   (self-loop term doubles as initializer)
__global__ void k_agg_init(const float* __restrict__ dinv, const float* __restrict__ h,
                           float* __restrict__ agg, int N) {
    int idx = blockIdx.x * blockDim.x + threadIdx.x;
    int i  = idx >> 5;
    if (i >= N) return;
    int fo = (idx & 31) * 4;
    float d  = dinv[i];
    float c  = d * d;
    float4 v = *(const float4*)(h + (size_t)i * FDIM + fo);
    float4 o = {c * v.x, c * v.y, c * v.z, c * v.w};
    *(float4*)(agg + (size_t)i * FDIM + fo) = o;
}

// agg[row[e],:] += dinv[row]*dinv[col] * h[col,:]  — 32 lanes per edge, float4/lane.
__global__ void k_agg_edges(const int* __restrict__ row, const int* __restrict__ col,
                            const float* __restrict__ dinv, const float* __restrict__ h,
                            float* __restrict__ agg, int E) {
    int idx = blockIdx.x * blockDim.x + threadIdx.x;
    int e = idx >> 5;
    if (e >= E) return;
    int fo = (idx & 31) * 4;

    // gfx1250 global_prefetch_b8: pull a future source row toward L2/WGP$
    if ((idx & 31) == 0) {
        int ep = e + 4096;
        if (ep < E) __builtin_prefetch(&h[(size_t)col[ep] * FDIM], 0, 1);
    }

    int r = row[e], c = col[e];
    float coef = dinv[r] * dinv[c];
    float4 v = *(const float4*)(h + (size_t)c * FDIM + fo);
    float* dst = agg + (size_t)r * FDIM + fo;
    atomicAdd(dst + 0, coef * v.x);
    atomicAdd(dst + 1, coef * v.y);
    atomicAdd(dst + 2, coef * v.z);
    atomicAdd(dst + 3, coef * v.w);
}

// ---------------- row-wise log_softmax, one wave32 per row ----------------
__global__ void k_log_softmax(float* __restrict__ out, int N) {
    int wid  = (blockIdx.x * blockDim.x + threadIdx.x) >> 5;
    int lane = threadIdx.x & 31;
    if (wid >= N) return;
    float* rowp = out + (size_t)wid * FDIM;

    float v[4];
    float m = -INFINITY;
#pragma unroll
    for (int j = 0; j < 4; ++j) {
        v[j] = rowp[lane + 32 * j];
        m = fmaxf(m, v[j]);
    }
#pragma unroll
    for (int off = 16; off > 0; off >>= 1)
        m = fmaxf(m, __shfl_xor(m, off, 32));

    float s = 0.0f;
#pragma unroll
    for (int j = 0; j < 4; ++j) s += __expf(v[j] - m);
#pragma unroll
    for (int off = 16; off > 0; off >>= 1)
        s += __shfl_xor(s, off, 32);

    float ls = __logf(s);
#pragma unroll
    for (int j = 0; j < 4; ++j) rowp[lane + 32 * j] = v[j] - m - ls;
}

// ---------------- launcher ----------------
extern "C" void kernel_launch(void* const* d_in, const int* in_sizes, int n_in,
                              void* d_out, int out_size, void* d_ws, size_t ws_size,
                              hipStream_t stream) {
    const float* x  = (const float*)d_in[0];
    const float* W0 = (const float*)d_in[1];
    const float* b0 = (const float*)d_in[2];
    const float* W1 = (const float*)d_in[3];
    const float* b1 = (const float*)d_in[4];
    const int*   row = (const int*)d_in[5];
    const int*   col = (const int*)d_in[6];

    const int N = in_sizes[0] / FDIM;
    const int E = in_sizes[5];
    float* outp = (float*)d_out;

    // workspace layout: dinv[N] | h[N*F] | agg1[N*F]   (~103 MB)
    float* dinv = (float*)d_ws;
    float* h    = dinv + (((size_t)N + 255) & ~(size_t)255);
    float* agg1 = h + (size_t)N * FDIM;

    const int nodeBlocks = (N + 255) / 256;
    const int edgeBlocks = (E + 255) / 256;
    const int gemmBlocks = (N + 63) / 64;
    const long long nodeF = (long long)N * 32;   // node x float4 lanes
    const long long edgeF = (long long)E * 32;   // edge x float4 lanes
    const unsigned nodeFBlocks = (unsigned)((nodeF + 255) / 256);
    const unsigned edgeFBlocks = (unsigned)((edgeF + 255) / 256);

    // degrees -> dinv
    k_init_deg<<<nodeBlocks, 256, 0, stream>>>(dinv, N);
    k_count_deg<<<edgeBlocks, 256, 0, stream>>>(col, dinv, E);
    k_finalize_dinv<<<nodeBlocks, 256, 0, stream>>>(dinv, N);

    // layer 0
    k_gemm_bias<false><<<gemmBlocks, 128, 0, stream>>>(x, W0, b0, h, N);
    k_agg_init<<<nodeFBlocks, 256, 0, stream>>>(dinv, h, agg1, N);
    k_agg_edges<<<edgeFBlocks, 256, 0, stream>>>(row, col, dinv, h, agg1, E);

    // layer 1 (ReLU folded into A-operand load)
    k_gemm_bias<true><<<gemmBlocks, 128, 0, stream>>>(agg1, W1, b1, h, N);
    k_agg_init<<<nodeFBlocks, 256, 0, stream>>>(dinv, h, outp, N);
    k_agg_edges<<<edgeFBlocks, 256, 0, stream>>>(row, col, dinv, h, outp, E);

    // log_softmax in place on d_out
    k_log_softmax<<<(N + 7) / 8, 256, 0, stream>>>(outp, N);
}